// TopK_10393820856567
// MI455X (gfx1250) — compile-verified
//
#include <hip/hip_runtime.h>
#include <hip/hip_bf16.h>

// Row-wise top-K masking: out[r, :] = x[r, :] * mask, mask keeps the K=512
// largest entries of each row. One workgroup per row; row staged in LDS via
// gfx1250 async global->LDS loads; exact threshold via 4-pass radix select
// on order-preserving uint keys; ties resolved with an LDS counter so exactly
// K elements are kept.

#define F_DIM    8192
#define K_SEL    512
#define THREADS  256
#define CHUNKS   (F_DIM / 4)   // float4 chunks per row = 2048

typedef int v4i __attribute__((ext_vector_type(4)));
typedef __attribute__((address_space(1))) v4i g_v4i;   // global AS
typedef __attribute__((address_space(3))) v4i l_v4i;   // LDS AS

__device__ __forceinline__ void async_copy_b128(const float* g, float* l) {
#if __has_builtin(__builtin_amdgcn_global_load_async_to_lds_b128)
    // Signature (probe-confirmed): (v4i AS1* src, v4i AS3* dst, Imm offset, Imm cpol)
    // Generic LDS pointers carry the LDS byte offset in the low 32 bits
    // (per the ISA aperture rule).
    __builtin_amdgcn_global_load_async_to_lds_b128(
        (g_v4i*)(unsigned long long)g,
        (l_v4i*)(unsigned int)(unsigned long long)l,
        0, 0);
#else
    unsigned int loff = (unsigned int)(unsigned long long)l;
    asm volatile("global_load_async_to_lds_b128 %0, %1, off"
                 :: "v"(loff), "v"(g) : "memory");
#endif
}

__device__ __forceinline__ void wait_async0() {
#if __has_builtin(__builtin_amdgcn_s_wait_asynccnt)
    __builtin_amdgcn_s_wait_asynccnt(0);
#else
    asm volatile("s_wait_asynccnt 0" ::: "memory");
#endif
}

// Order-preserving map: larger float -> larger uint (total order, -0 < +0).
__device__ __forceinline__ unsigned int fkey(float f) {
    unsigned int u = __float_as_uint(f);
    return (u & 0x80000000u) ? ~u : (u | 0x80000000u);
}

__global__ __launch_bounds__(THREADS)
void topk_mask_kernel(const float* __restrict__ x, float* __restrict__ out) {
    __shared__ __align__(16) float row[F_DIM];       // 32 KB row stage
    __shared__ unsigned int hist[256];
    __shared__ unsigned int sPrefVal, sPrefMask, sNeed, sTie;

    const int r   = blockIdx.x;
    const int tid = threadIdx.x;
    const float* __restrict__ src = x   + (size_t)r * F_DIM;
    float*       __restrict__ dst = out + (size_t)r * F_DIM;

    // ---- Stage the whole row into LDS with async b128 copies (ASYNCcnt) ----
    for (int c = tid; c < CHUNKS; c += THREADS) {
        async_copy_b128(src + 4 * c, &row[4 * c]);
    }
    wait_async0();          // drain this wave's async loads
    __syncthreads();        // all waves' data visible in LDS

    if (tid == 0) { sPrefVal = 0u; sPrefMask = 0u; sNeed = K_SEL; sTie = 0u; }

    // ---- 4-pass radix select (MSB -> LSB) for the K-th largest key ----
    for (int shift = 24; shift >= 0; shift -= 8) {
        if (tid < 256) hist[tid] = 0u;
        __syncthreads();                       // hist zeroed; prefix visible
        const unsigned int pv = sPrefVal;
        const unsigned int pm = sPrefMask;
        for (int c = tid; c < CHUNKS; c += THREADS) {
            float4 v = *(const float4*)&row[4 * c];
            unsigned int k0 = fkey(v.x), k1 = fkey(v.y);
            unsigned int k2 = fkey(v.z), k3 = fkey(v.w);
            if ((k0 & pm) == pv) atomicAdd(&hist[(k0 >> shift) & 255u], 1u);
            if ((k1 & pm) == pv) atomicAdd(&hist[(k1 >> shift) & 255u], 1u);
            if ((k2 & pm) == pv) atomicAdd(&hist[(k2 >> shift) & 255u], 1u);
            if ((k3 & pm) == pv) atomicAdd(&hist[(k3 >> shift) & 255u], 1u);
        }
        __syncthreads();
        if (tid == 0) {
            unsigned int need = sNeed, cum = 0u;
            int sel = 0;
            for (int b = 255; b >= 0; --b) {
                unsigned int cb = hist[b];
                if (cum + cb >= need) { sel = b; sNeed = need - cum; break; }
                cum += cb;
            }
            sPrefVal  = pv | ((unsigned int)sel << shift);
            sPrefMask = pm | (255u << shift);
        }
        __syncthreads();
    }

    const unsigned int T    = sPrefVal;  // exact 32-bit key of the K-th largest
    const unsigned int need = sNeed;     // how many elements equal to T to keep

    // ---- Masked write-back: key > T kept; key == T kept for first `need` ----
    for (int c = tid; c < CHUNKS; c += THREADS) {
        float4 v = *(const float4*)&row[4 * c];
        float in[4] = { v.x, v.y, v.z, v.w };
        float o[4];
#pragma unroll
        for (int j = 0; j < 4; ++j) {
            unsigned int k = fkey(in[j]);
            float rout = 0.0f;
            if (k > T) {
                rout = in[j];
            } else if (k == T) {
                unsigned int pos = atomicAdd(&sTie, 1u);
                if (pos < need) rout = in[j];
            }
            o[j] = rout;
        }
        float4 ov; ov.x = o[0]; ov.y = o[1]; ov.z = o[2]; ov.w = o[3];
        *(float4*)(dst + 4 * c) = ov;
    }
}

extern "C" void kernel_launch(void* const* d_in, const int* in_sizes, int n_in,
                              void* d_out, int out_size, void* d_ws, size_t ws_size,
                              hipStream_t stream) {
    const float* x = (const float*)d_in[0];
    float* out     = (float*)d_out;
    const int rows = in_sizes[0] / F_DIM;   // 8192
    topk_mask_kernel<<<rows, THREADS, 0, stream>>>(x, out);
}